// DirectionalGNN_41721312314262
// MI455X (gfx1250) — compile-verified
//
#include <hip/hip_runtime.h>
#include <hip/hip_bf16.h>
#include <math.h>

// Problem constants (from reference)
#define NN 100000
#define EE 1000000
#define GG 1024
#define HH 64
#define KE 72      // H + EA
#define FCC 128
#define BW 7

typedef float v2f __attribute__((ext_vector_type(2)));
typedef float v8f __attribute__((ext_vector_type(8)));

__device__ __forceinline__ void atomAddF(float* p, float v) {
  unsafeAtomicAdd(p, v);  // hardware global_atomic_add_f32
}

// ---------------- utility kernels ----------------

__global__ __launch_bounds__(256) void zero_kernel(float* __restrict__ p, int n) {
  int i = blockIdx.x * 256 + threadIdx.x;
  if (i < n) p[i] = 0.0f;
}

__global__ __launch_bounds__(256) void degree_kernel(const int* __restrict__ dst,
                                                     const int* __restrict__ batch,
                                                     float* __restrict__ deg,
                                                     float* __restrict__ gcnt) {
  int i = blockIdx.x * 256 + threadIdx.x;
  if (i < EE) atomAddF(&deg[dst[i]], 1.0f);
  if (i < NN) atomAddF(&gcnt[batch[i]], 1.0f);
}

__global__ __launch_bounds__(256) void in_proj_kernel(const float* __restrict__ x,
                                                      const float* __restrict__ W,
                                                      const float* __restrict__ b,
                                                      float* __restrict__ h) {
  int i = blockIdx.x * 256 + threadIdx.x;
  if (i >= NN * HH) return;
  int n = i >> 6, c = i & 63;
  float acc = b[c] + x[n * 3 + 0] * W[c] + x[n * 3 + 1] * W[64 + c] + x[n * 3 + 2] * W[128 + c];
  h[i] = fmaxf(acc, 0.0f);
}

// ---------------- edge MLP (fused gather + 2 GEMMs + scatter) ----------------
// 128 threads = 4 waves, each wave owns a 16-edge tile.
// WMMA f32 16x16x4:
//   A (16x4): lanes 0-15 hold M=lane K={k0,k0+1}; lanes 16-31 hold M=lane-16 K={k0+2,k0+3}
//   B (4x16): symmetric (lane half selects K pair, lane&15 selects N)
//   C/D: VGPR r -> row r (lanes 0-15) / row r+8 (lanes 16-31), col = lane&15

__global__ __launch_bounds__(128) void edge_mlp_kernel(
    const float* __restrict__ h, const float* __restrict__ ea,
    const int* __restrict__ src, const int* __restrict__ dst,
    const float* __restrict__ W1, const float* __restrict__ b1,
    const float* __restrict__ W2, const float* __restrict__ b2,
    float* __restrict__ aggr) {
  __shared__ float sW1[64 * 76];      // W1 transposed: [out_c][k], pitch 76 (bank-conflict free)
  __shared__ float sW2[64 * 68];      // W2 transposed: [out_c][k], pitch 68
  __shared__ float sB1[64];
  __shared__ float sB2[64];
  __shared__ float sA[4][16 * 76];    // per-wave activation tile [16][72], pitch 76

  const int tid = threadIdx.x;
  for (int i = tid; i < KE * HH; i += 128) {
    int k = i >> 6, c = i & 63;
    sW1[c * 76 + k] = W1[i];
  }
  for (int i = tid; i < HH * HH; i += 128) {
    int k = i >> 6, c = i & 63;
    sW2[c * 68 + k] = W2[i];
  }
  if (tid < 64) { sB1[tid] = b1[tid]; sB2[tid] = b2[tid]; }
  __syncthreads();

  const int wave = tid >> 5;
  const int lane = tid & 31;
  const int m = lane & 15;
  const int kh = (lane >> 4) << 1;      // 0 or 2
  const int rowoff = (lane >> 4) << 3;  // 0 or 8 (C/D row offset)
  const int ebase = blockIdx.x * 64 + wave * 16;
  float* A = sA[wave];

  // gather h[src[e]] rows: 16 rows x 16 float4 -> 8 float4 per lane
#pragma unroll
  for (int it = 0; it < 8; ++it) {
    int i = lane + it * 32;
    int r = i >> 4, q = i & 15;
    int e = ebase + r; if (e >= EE) e = EE - 1;   // clamp (keep EXEC full)
    const float4 v = ((const float4*)(h + (size_t)src[e] * HH))[q];
    float* d = &A[r * 76 + q * 4];
    d[0] = v.x; d[1] = v.y; d[2] = v.z; d[3] = v.w;
  }
  // gather edge_attr: 16 rows x 2 float4 -> 1 per lane
  {
    int r = lane >> 1, q = lane & 1;
    int e = ebase + r; if (e >= EE) e = EE - 1;
    const float4 v = ((const float4*)(ea + (size_t)e * 8))[q];
    float* d = &A[r * 76 + 64 + q * 4];
    d[0] = v.x; d[1] = v.y; d[2] = v.z; d[3] = v.w;
  }
  __syncthreads();

  // GEMM1: [16,72] @ [72,64] + b1, ReLU
  v8f acc1[4];
#pragma unroll
  for (int ct = 0; ct < 4; ++ct) {
    float bv = sB1[ct * 16 + m];
    v8f acc = {bv, bv, bv, bv, bv, bv, bv, bv};
    const float* wcol = &sW1[(ct * 16 + m) * 76];
#pragma unroll
    for (int k0 = 0; k0 < KE; k0 += 4) {
      v2f a, b;
      a.x = A[m * 76 + k0 + kh]; a.y = A[m * 76 + k0 + kh + 1];
      b.x = wcol[k0 + kh];       b.y = wcol[k0 + kh + 1];
      acc = __builtin_amdgcn_wmma_f32_16x16x4_f32(false, a, false, b, (short)0, acc, false, false);
    }
    acc1[ct] = acc;
  }
  // ReLU result back into wave-local A tile (LDS is in-order within a wave)
#pragma unroll
  for (int ct = 0; ct < 4; ++ct)
#pragma unroll
    for (int r = 0; r < 8; ++r)
      A[(r + rowoff) * 76 + ct * 16 + m] = fmaxf(acc1[ct][r], 0.0f);

  // GEMM2: [16,64] @ [64,64] + b2
  v8f acc2[4];
#pragma unroll
  for (int ct = 0; ct < 4; ++ct) {
    float bv = sB2[ct * 16 + m];
    v8f acc = {bv, bv, bv, bv, bv, bv, bv, bv};
    const float* wcol = &sW2[(ct * 16 + m) * 68];
#pragma unroll
    for (int k0 = 0; k0 < HH; k0 += 4) {
      v2f a, b;
      a.x = A[m * 76 + k0 + kh]; a.y = A[m * 76 + k0 + kh + 1];
      b.x = wcol[k0 + kh];       b.y = wcol[k0 + kh + 1];
      acc = __builtin_amdgcn_wmma_f32_16x16x4_f32(false, a, false, b, (short)0, acc, false, false);
    }
    acc2[ct] = acc;
  }

  // scatter-add into aggr[dst[e]] (inv_deg applied later in node kernel)
#pragma unroll
  for (int r = 0; r < 8; ++r) {
    int e = ebase + r + rowoff;
    if (e < EE) {
      float* out = aggr + (size_t)dst[e] * HH;
#pragma unroll
      for (int ct = 0; ct < 4; ++ct)
        atomAddF(out + ct * 16 + m, acc2[ct][r]);
    }
  }
}

// ---------------- node update: h = relu(aggr*inv_deg + h@Wn + bn) ----------------
__global__ __launch_bounds__(256) void node_update_kernel(
    const float* __restrict__ h, const float* __restrict__ aggr,
    const float* __restrict__ deg,
    const float* __restrict__ Wn, const float* __restrict__ bn,
    float* __restrict__ hout) {
  __shared__ float sW[64 * 68];
  __shared__ float sB[64];
  __shared__ float sA[8][16 * 68];

  const int tid = threadIdx.x;
  for (int i = tid; i < HH * HH; i += 256) {
    int k = i >> 6, c = i & 63;
    sW[c * 68 + k] = Wn[i];
  }
  if (tid < 64) sB[tid] = bn[tid];
  __syncthreads();

  const int wave = tid >> 5, lane = tid & 31;
  const int m = lane & 15, kh = (lane >> 4) << 1, rowoff = (lane >> 4) << 3;
  const int nbase = blockIdx.x * 128 + wave * 16;
  float* A = sA[wave];

#pragma unroll
  for (int it = 0; it < 8; ++it) {
    int i = lane + it * 32;
    int r = i >> 4, q = i & 15;
    int n = nbase + r; if (n >= NN) n = NN - 1;
    const float4 v = ((const float4*)(h + (size_t)n * HH))[q];
    float* d = &A[r * 68 + q * 4];
    d[0] = v.x; d[1] = v.y; d[2] = v.z; d[3] = v.w;
  }
  __syncthreads();

  v8f accv[4];
#pragma unroll
  for (int ct = 0; ct < 4; ++ct) {
    float bv = sB[ct * 16 + m];
    v8f acc = {bv, bv, bv, bv, bv, bv, bv, bv};
    const float* wcol = &sW[(ct * 16 + m) * 68];
#pragma unroll
    for (int k0 = 0; k0 < HH; k0 += 4) {
      v2f a, b;
      a.x = A[m * 68 + k0 + kh]; a.y = A[m * 68 + k0 + kh + 1];
      b.x = wcol[k0 + kh];       b.y = wcol[k0 + kh + 1];
      acc = __builtin_amdgcn_wmma_f32_16x16x4_f32(false, a, false, b, (short)0, acc, false, false);
    }
    accv[ct] = acc;
  }

#pragma unroll
  for (int r = 0; r < 8; ++r) {
    int n = nbase + r + rowoff;
    if (n < NN) {
      float inv = 1.0f / fmaxf(deg[n], 1.0f);
#pragma unroll
      for (int ct = 0; ct < 4; ++ct) {
        int c = ct * 16 + m;
        float v = accv[ct][r] + aggr[(size_t)n * HH + c] * inv;
        hout[(size_t)n * HH + c] = fmaxf(v, 0.0f);
      }
    }
  }
}

// ---------------- pooling + head ----------------
__global__ __launch_bounds__(256) void pool_kernel(const float* __restrict__ h,
                                                   const int* __restrict__ batch,
                                                   float* __restrict__ gsum) {
  int i = blockIdx.x * 256 + threadIdx.x;
  if (i >= NN * HH) return;
  int n = i >> 6, c = i & 63;
  atomAddF(&gsum[(size_t)batch[n] * HH + c], h[i]);
}

__global__ __launch_bounds__(128) void head_kernel(
    const float* __restrict__ gsum, const float* __restrict__ gcnt,
    const float* __restrict__ Wfc, const float* __restrict__ bfc,
    const float* __restrict__ Wp, const float* __restrict__ bp,
    const float* __restrict__ Wv, const float* __restrict__ bv,
    float* __restrict__ out_policy, float* __restrict__ out_value) {
  int g = blockIdx.x, t = threadIdx.x;
  __shared__ float sg[64];
  __shared__ float sf[128];
  if (t < 64) {
    float inv = 1.0f / fmaxf(gcnt[g], 1.0f);
    sg[t] = gsum[(size_t)g * HH + t] * inv;
  }
  __syncthreads();
  float acc = bfc[t];
#pragma unroll 8
  for (int k = 0; k < 64; ++k) acc += sg[k] * Wfc[k * FCC + t];
  sf[t] = fmaxf(acc, 0.0f);
  __syncthreads();
  if (t < BW) {
    float a = bp[t];
#pragma unroll 8
    for (int k = 0; k < FCC; ++k) a += sf[k] * Wp[k * BW + t];
    out_policy[g * BW + t] = a;
  } else if (t == BW) {
    float a = bv[0];
#pragma unroll 8
    for (int k = 0; k < FCC; ++k) a += sf[k] * Wv[k];
    out_value[g] = tanhf(a);
  }
}

// ---------------- launch ----------------
extern "C" void kernel_launch(void* const* d_in, const int* in_sizes, int n_in,
                              void* d_out, int out_size, void* d_ws, size_t ws_size,
                              hipStream_t stream) {
  (void)in_sizes; (void)n_in; (void)out_size; (void)ws_size;
  // setup_inputs() insertion order, params flattened depth-first:
  const float* x   = (const float*)d_in[0];    // [N,3]
  const float* ea  = (const float*)d_in[1];    // [E,8]
  const int*   ei  = (const int*)d_in[2];      // [2,E]
  const int*   bat = (const int*)d_in[3];      // [N]
  const float* Win = (const float*)d_in[4];    // [3,64]
  const float* bin = (const float*)d_in[5];    // [64]
  const float* W1[2] = {(const float*)d_in[6],  (const float*)d_in[12]};
  const float* b1[2] = {(const float*)d_in[7],  (const float*)d_in[13]};
  const float* W2[2] = {(const float*)d_in[8],  (const float*)d_in[14]};
  const float* b2[2] = {(const float*)d_in[9],  (const float*)d_in[15]};
  const float* Wn[2] = {(const float*)d_in[10], (const float*)d_in[16]};
  const float* bn[2] = {(const float*)d_in[11], (const float*)d_in[17]};
  const float* Wfc = (const float*)d_in[18];   // [64,128]
  const float* bfc = (const float*)d_in[19];   // [128]
  const float* Wp  = (const float*)d_in[20];   // [128,7]
  const float* bp  = (const float*)d_in[21];   // [7]
  const float* Wv  = (const float*)d_in[22];   // [128,1]
  const float* bv  = (const float*)d_in[23];   // [1]

  const int* srcI = ei;
  const int* dstI = ei + EE;

  // workspace layout (floats): h0 | h1 | aggr | deg | gsum | gcnt
  float* h0   = (float*)d_ws;
  float* h1   = h0 + (size_t)NN * HH;
  float* aggr = h1 + (size_t)NN * HH;
  float* deg  = aggr + (size_t)NN * HH;
  float* gsum = deg + NN;
  float* gcnt = gsum + (size_t)GG * HH;

  // zero aggr..gcnt (contiguous region)
  const int zc = NN * HH + NN + GG * HH + GG;
  zero_kernel<<<(zc + 255) / 256, 256, 0, stream>>>(aggr, zc);
  degree_kernel<<<(EE + 255) / 256, 256, 0, stream>>>(dstI, bat, deg, gcnt);
  in_proj_kernel<<<(NN * HH + 255) / 256, 256, 0, stream>>>(x, Win, bin, h0);

  // layer 0: h0 -> h1
  edge_mlp_kernel<<<(EE + 63) / 64, 128, 0, stream>>>(h0, ea, srcI, dstI,
                                                      W1[0], b1[0], W2[0], b2[0], aggr);
  node_update_kernel<<<(NN + 127) / 128, 256, 0, stream>>>(h0, aggr, deg, Wn[0], bn[0], h1);

  // layer 1: h1 -> h0
  zero_kernel<<<(NN * HH + 255) / 256, 256, 0, stream>>>(aggr, NN * HH);
  edge_mlp_kernel<<<(EE + 63) / 64, 128, 0, stream>>>(h1, ea, srcI, dstI,
                                                      W1[1], b1[1], W2[1], b2[1], aggr);
  node_update_kernel<<<(NN + 127) / 128, 256, 0, stream>>>(h1, aggr, deg, Wn[1], bn[1], h0);

  // pool + head
  pool_kernel<<<(NN * HH + 255) / 256, 256, 0, stream>>>(h0, bat, gsum);
  head_kernel<<<GG, 128, 0, stream>>>(gsum, gcnt, Wfc, bfc, Wp, bp, Wv, bv,
                                      (float*)d_out, (float*)d_out + (size_t)GG * BW);
}